// QuadrotorControllerModule_41961830481953
// MI455X (gfx1250) — compile-verified
//
#include <hip/hip_runtime.h>
#include <hip/hip_bf16.h>

#ifndef __has_builtin
#define __has_builtin(x) 0
#endif

#if __has_builtin(__builtin_amdgcn_global_load_async_to_lds_b128)
#define USE_ASYNC_LDS 1
#else
#define USE_ASYNC_LDS 0
#endif

#if __has_builtin(__builtin_amdgcn_s_wait_asynccnt)
#define WAIT_ASYNC(n) __builtin_amdgcn_s_wait_asynccnt(n)
#else
#define WAIT_ASYNC(n) asm volatile("s_wait_asynccnt %0" ::"i"(n) : "memory")
#endif

#define GLOBAL_AS __attribute__((address_space(1)))
#define LDS_AS __attribute__((address_space(3)))

typedef int v4i __attribute__((__vector_size__(4 * sizeof(int))));

static constexpr int T_STEPS = 500;
static constexpr int BATCH = 4096;
static constexpr float DT_C = 0.002f;
static constexpr float G_C = 9.81f;
static constexpr double ARM_D = 0.707106781 * 0.046;
static constexpr float ARM_C = (float)ARM_D;
static constexpr float T2T_C = 0.006f;
static constexpr float INV4A = (float)(1.0 / (4.0 * ARM_D));  // B0inv analytic
static constexpr float INV4T = (float)(1.0 / (4.0 * 0.006));

__global__ __launch_bounds__(32) void quad_rollout_kernel(
    const float* __restrict__ sp_all, const float* __restrict__ kp,
    const float* __restrict__ kv, const float* __restrict__ kw,
    const float* __restrict__ kr, const float* __restrict__ mass,
    const float* __restrict__ inertia, float* __restrict__ out_ys,
    float* __restrict__ out_rd, float* __restrict__ out_dw) {
  const int lane = threadIdx.x;
  const int b = blockIdx.x * 32 + lane;

  // Uniform parameters -> scalar loads
  const float kp0 = kp[0], kp1 = kp[1], kp2 = kp[2];
  const float kv0 = kv[0], kv1 = kv[1], kv2 = kv[2];
  const float kw0 = kw[0], kw1 = kw[1], kw2 = kw[2];
  const float kr0 = kr[0], kr1 = kr[1], kr2 = kr[2];
  const float m = mass[0];
  const float inv_m = 1.0f / m;
  const float I00 = inertia[0], I01 = inertia[1], I02 = inertia[2];
  const float I10 = inertia[3], I11 = inertia[4], I12 = inertia[5];
  const float I20 = inertia[6], I21 = inertia[7], I22 = inertia[8];
  // 3x3 inverse (adjugate), once per thread
  const float cf0 = I11 * I22 - I12 * I21;
  const float cf1 = I12 * I20 - I10 * I22;
  const float cf2 = I10 * I21 - I11 * I20;
  const float invdet = 1.0f / (I00 * cf0 + I01 * cf1 + I02 * cf2);
  const float Ii00 = cf0 * invdet;
  const float Ii01 = (I02 * I21 - I01 * I22) * invdet;
  const float Ii02 = (I01 * I12 - I02 * I11) * invdet;
  const float Ii10 = cf1 * invdet;
  const float Ii11 = (I00 * I22 - I02 * I20) * invdet;
  const float Ii12 = (I02 * I10 - I00 * I12) * invdet;
  const float Ii20 = cf2 * invdet;
  const float Ii21 = (I01 * I20 - I00 * I21) * invdet;
  const float Ii22 = (I00 * I11 - I01 * I10) * invdet;

  // Initial state from setpoints[0, b, 0:6], q = (1,0,0,0), w = 0
  const float* s0 = sp_all + (size_t)b * 13u;
  float px = s0[0], py = s0[1], pz = s0[2];
  float vx = s0[3], vy = s0[4], vz = s0[5];
  float qw = 1.f, qx = 0.f, qy = 0.f, qz = 0.f;
  float w0 = 0.f, w1 = 0.f, w2 = 0.f;

#if USE_ASYNC_LDS
  // Double-buffered async staging: per step this wave consumes 32*13 floats
  // = 1664 B = 104 x b128 chunks (16B aligned in both global and LDS).
  __shared__ __align__(16) float spbuf[2][32 * 13];
  const char* gblk0 = (const char*)(sp_all + (size_t)blockIdx.x * 32u * 13u);
  auto stage = [&](int t, int bufi) {
    const char* g = gblk0 + (size_t)t * (size_t)BATCH * 13u * 4u;
    char* l = (char*)&spbuf[bufi][0];
#pragma unroll
    for (int r = 0; r < 3; ++r) {
      const int off = (lane + 32 * r) * 16;
      __builtin_amdgcn_global_load_async_to_lds_b128(
          (GLOBAL_AS v4i*)(g + off), (LDS_AS v4i*)(l + off), 0, 0);
    }
    {
      const int off = (96 + lane) * 16;
      if (lane < 8)
        __builtin_amdgcn_global_load_async_to_lds_b128(
            (GLOBAL_AS v4i*)(g + off), (LDS_AS v4i*)(l + off), 0, 0);
    }
  };
  stage(0, 0);
#endif

#pragma unroll 1
  for (int t = 0; t < T_STEPS; ++t) {
    float sp[13];
#if USE_ASYNC_LDS
    if (t + 1 < T_STEPS) {
      stage(t + 1, (t + 1) & 1);  // prefetch next step, 4 async ops outstanding
      WAIT_ASYNC(4);              // wait for current step's 4 (in-order)
    } else {
      WAIT_ASYNC(0);
    }
    asm volatile("" ::: "memory");
    const float* sl = &spbuf[t & 1][lane * 13];  // stride 13: bank-conflict free
#pragma unroll
    for (int k = 0; k < 13; ++k) sp[k] = sl[k];
#else
    const float* sg = sp_all + ((size_t)t * BATCH + (size_t)b) * 13u;
#pragma unroll
    for (int k = 0; k < 13; ++k) sp[k] = __builtin_nontemporal_load(sg + k);
#endif

    const size_t idx = (size_t)t * BATCH + (size_t)b;
    // ys[t] = state at step entry (streamed, never re-read -> NT stores)
    {
      float* y = out_ys + idx * 13u;
      __builtin_nontemporal_store(px, y + 0);
      __builtin_nontemporal_store(py, y + 1);
      __builtin_nontemporal_store(pz, y + 2);
      __builtin_nontemporal_store(vx, y + 3);
      __builtin_nontemporal_store(vy, y + 4);
      __builtin_nontemporal_store(vz, y + 5);
      __builtin_nontemporal_store(qw, y + 6);
      __builtin_nontemporal_store(qx, y + 7);
      __builtin_nontemporal_store(qy, y + 8);
      __builtin_nontemporal_store(qz, y + 9);
      __builtin_nontemporal_store(w0, y + 10);
      __builtin_nontemporal_store(w1, y + 11);
      __builtin_nontemporal_store(w2, y + 12);
    }

    const float pd0 = sp[0], pd1 = sp[1], pd2 = sp[2];
    const float vd0 = sp[3], vd1 = sp[4], vd2 = sp[5];
    const float ad0 = sp[6], ad1 = sp[7], ad2 = sp[8];
    const float jd0 = sp[9], jd1 = sp[10], jd2 = sp[11];
    const float yaw = sp[12];

    // Rotation matrix from quaternion
    const float xx = qx * qx, yy = qy * qy, zz = qz * qz;
    const float xy = qx * qy, xz = qx * qz, yz = qy * qz;
    const float qwx = qw * qx, qwy = qw * qy, qwz = qw * qz;
    const float R00 = 1.f - 2.f * (yy + zz), R01 = 2.f * (xy - qwz), R02 = 2.f * (xz + qwy);
    const float R10 = 2.f * (xy + qwz), R11 = 1.f - 2.f * (xx + zz), R12 = 2.f * (yz - qwx);
    const float R20 = 2.f * (xz - qwy), R21 = 2.f * (yz + qwx), R22 = 1.f - 2.f * (xx + yy);

    // Desired force
    const float ep0 = px - pd0, ep1 = py - pd1, ep2 = pz - pd2;
    const float ev0 = vx - vd0, ev1 = vy - vd1, ev2 = vz - vd2;
    const float Fd0 = m * (ad0 - kp0 * ep0 - kv0 * ev0);
    const float Fd1 = m * (ad1 - kp1 * ep1 - kv1 * ev1);
    const float Fd2 = m * (ad2 + G_C - kp2 * ep2 - kv2 * ev2);
    const float thrust = Fd0 * R02 + Fd1 * R12 + Fd2 * R22;
    const float inv_fn = 1.f / sqrtf(Fd0 * Fd0 + Fd1 * Fd1 + Fd2 * Fd2);
    const float zd0 = Fd0 * inv_fn, zd1 = Fd1 * inv_fn, zd2 = Fd2 * inv_fn;

    // Desired attitude
    const float cy = __cosf(yaw), sy = __sinf(yaw);
    const float cr0 = -zd2 * sy;
    const float cr1 = zd2 * cy;
    const float cr2 = zd0 * sy - zd1 * cy;
    const float inv_cn = 1.f / sqrtf(cr0 * cr0 + cr1 * cr1 + cr2 * cr2);
    const float yd0 = cr0 * inv_cn, yd1 = cr1 * inv_cn, yd2 = cr2 * inv_cn;
    const float xd0 = yd1 * zd2 - yd2 * zd1;
    const float xd1 = yd2 * zd0 - yd0 * zd2;
    const float xd2 = yd0 * zd1 - yd1 * zd0;

    // Desired angular rate (projected jerk)
    const float mz = m / thrust;
    const float zjd = zd0 * jd0 + zd1 * jd1 + zd2 * jd2;
    const float hw0 = mz * (jd0 - zjd * zd0);
    const float hw1 = mz * (jd1 - zjd * zd1);
    const float hw2 = mz * (jd2 - zjd * zd2);
    const float dw0 = -(hw0 * yd0 + hw1 * yd1 + hw2 * yd2);
    const float dw1 = hw0 * xd0 + hw1 * xd1 + hw2 * xd2;

    // Attitude error: M = Rd^T R, eR = 0.5 * vee(M - M^T)
    const float M01 = xd0 * R01 + xd1 * R11 + xd2 * R21;
    const float M02 = xd0 * R02 + xd1 * R12 + xd2 * R22;
    const float M10 = yd0 * R00 + yd1 * R10 + yd2 * R20;
    const float M12 = yd0 * R02 + yd1 * R12 + yd2 * R22;
    const float M20 = zd0 * R00 + zd1 * R10 + zd2 * R20;
    const float M21 = zd0 * R01 + zd1 * R11 + zd2 * R21;
    const float eR0 = 0.5f * (M21 - M12);
    const float eR1 = 0.5f * (M02 - M20);
    const float eR2 = 0.5f * (M10 - M01);

    // wd_body = R^T (Rd @ desW), desW_z = 0
    const float u0 = dw0 * xd0 + dw1 * yd0;
    const float u1 = dw0 * xd1 + dw1 * yd1;
    const float u2 = dw0 * xd2 + dw1 * yd2;
    const float wb0 = R00 * u0 + R10 * u1 + R20 * u2;
    const float wb1 = R01 * u0 + R11 * u1 + R21 * u2;
    const float wb2 = R02 * u0 + R12 * u1 + R22 * u2;
    const float ew0 = w0 - wb0, ew1 = w1 - wb1, ew2 = w2 - wb2;

    // Gyroscopic term: cross(w, I @ w)
    const float Iw0 = I00 * w0 + I01 * w1 + I02 * w2;
    const float Iw1 = I10 * w0 + I11 * w1 + I12 * w2;
    const float Iw2 = I20 * w0 + I21 * w1 + I22 * w2;
    const float cw0 = w1 * Iw2 - w2 * Iw1;
    const float cw1 = w2 * Iw0 - w0 * Iw2;
    const float cw2 = w0 * Iw1 - w1 * Iw0;
    const float tau0 = -kr0 * eR0 - kw0 * ew0 + cw0;
    const float tau1 = -kr1 * eR1 - kw1 * ew1 + cw1;
    const float tau2 = -kr2 * eR2 - kw2 * ew2 + cw2;

    // Mixer: force = B0inv @ eta ; eta2 = B0 @ force (analytic B0inv)
    const float hq = 0.25f * thrust;
    const float q1a = INV4A * tau0, q2a = INV4A * tau1, q3t = INV4T * tau2;
    const float f0 = hq - q1a - q2a - q3t;
    const float f1 = hq - q1a + q2a + q3t;
    const float f2 = hq + q1a + q2a - q3t;
    const float f3 = hq + q1a - q2a + q3t;
    const float fT = f0 + f1 + f2 + f3;
    const float tq0 = ARM_C * (-f0 - f1 + f2 + f3);
    const float tq1 = ARM_C * (-f0 + f1 + f2 - f3);
    const float tq2 = T2T_C * (-f0 + f1 - f2 + f3);

    // Store Rd (columns xd, yd, zd) and desW (streamed -> NT stores)
    {
      float* r = out_rd + idx * 9u;
      __builtin_nontemporal_store(xd0, r + 0);
      __builtin_nontemporal_store(yd0, r + 1);
      __builtin_nontemporal_store(zd0, r + 2);
      __builtin_nontemporal_store(xd1, r + 3);
      __builtin_nontemporal_store(yd1, r + 4);
      __builtin_nontemporal_store(zd1, r + 5);
      __builtin_nontemporal_store(xd2, r + 6);
      __builtin_nontemporal_store(yd2, r + 7);
      __builtin_nontemporal_store(zd2, r + 8);
      float* d = out_dw + idx * 3u;
      __builtin_nontemporal_store(dw0, d + 0);
      __builtin_nontemporal_store(dw1, d + 1);
      __builtin_nontemporal_store(0.0f, d + 2);
    }

    // Integrate dynamics
    const float fom = fT * inv_m;
    const float acc0 = R02 * fom;
    const float acc1 = R12 * fom;
    const float acc2 = R22 * fom - G_C;
    const float a0 = tq0 - cw0, a1 = tq1 - cw1, a2 = tq2 - cw2;
    const float dwx = Ii00 * a0 + Ii01 * a1 + Ii02 * a2;
    const float dwy = Ii10 * a0 + Ii11 * a1 + Ii12 * a2;
    const float dwz = Ii20 * a0 + Ii21 * a1 + Ii22 * a2;

    const float qd0 = -0.5f * (qx * w0 + qy * w1 + qz * w2);
    const float qd1 = 0.5f * (qw * w0 + qy * w2 - qz * w1);
    const float qd2 = 0.5f * (qw * w1 + qz * w0 - qx * w2);
    const float qd3 = 0.5f * (qw * w2 + qx * w1 - qy * w0);
    const float nqw = qw + DT_C * qd0;
    const float nqx = qx + DT_C * qd1;
    const float nqy = qy + DT_C * qd2;
    const float nqz = qz + DT_C * qd3;
    const float inv_qn =
        1.f / sqrtf(nqw * nqw + nqx * nqx + nqy * nqy + nqz * nqz);

    px += DT_C * vx;
    py += DT_C * vy;
    pz += DT_C * vz;
    vx += DT_C * acc0;
    vy += DT_C * acc1;
    vz += DT_C * acc2;
    w0 += DT_C * dwx;
    w1 += DT_C * dwy;
    w2 += DT_C * dwz;
    qw = nqw * inv_qn;
    qx = nqx * inv_qn;
    qy = nqy * inv_qn;
    qz = nqz * inv_qn;
  }
}

extern "C" void kernel_launch(void* const* d_in, const int* in_sizes, int n_in,
                              void* d_out, int out_size, void* d_ws,
                              size_t ws_size, hipStream_t stream) {
  const float* sp = (const float*)d_in[0];
  const float* kp = (const float*)d_in[1];
  const float* kv = (const float*)d_in[2];
  const float* kw = (const float*)d_in[3];
  const float* kr = (const float*)d_in[4];
  const float* mass = (const float*)d_in[5];
  const float* inertia = (const float*)d_in[6];
  float* out = (float*)d_out;
  float* out_ys = out;                                     // (T,B,13)
  float* out_rd = out_ys + (size_t)T_STEPS * BATCH * 13u;  // (T,B,3,3)
  float* out_dw = out_rd + (size_t)T_STEPS * BATCH * 9u;   // (T,B,3)
  // One lane per batch element; 32-thread (1-wave) blocks -> 128 blocks spread
  // across WGPs, wave-synchronous async-LDS double buffering (no barriers).
  quad_rollout_kernel<<<BATCH / 32, 32, 0, stream>>>(
      sp, kp, kv, kw, kr, mass, inertia, out_ys, out_rd, out_dw);
}